// GCNLayer_76690936037554
// MI455X (gfx1250) — compile-verified
//
#include <hip/hip_runtime.h>

typedef __attribute__((ext_vector_type(2))) float v2f;
typedef __attribute__((ext_vector_type(8))) float v8f;

// ---------------------------------------------------------------------------
// Kernel 0: zero the workspace region (agg[N*128] ++ deg[N], contiguous)
// ---------------------------------------------------------------------------
__global__ void __launch_bounds__(256)
gcn_zero_kernel(float* __restrict__ p, long long n) {
    long long i = (long long)blockIdx.x * blockDim.x + threadIdx.x;
    long long stride = (long long)gridDim.x * blockDim.x;
    for (; i < n; i += stride) p[i] = 0.0f;
}

// ---------------------------------------------------------------------------
// Kernel 1: edge aggregation.
//   agg[dst] += x[src] * w_e      (exploits linearity: aggregate BEFORE GEMM)
//   deg[dst] += 1
// One edge per 32 threads; each lane handles 4 consecutive floats (float4).
// ---------------------------------------------------------------------------
__global__ void __launch_bounds__(256)
gcn_aggregate_kernel(const float* __restrict__ x,
                     const long long* __restrict__ src,
                     const long long* __restrict__ dst,
                     const float* __restrict__ ew,
                     float* __restrict__ agg,
                     float* __restrict__ deg,
                     long long E) {
    long long t = (long long)blockIdx.x * blockDim.x + threadIdx.x;
    long long e = t >> 5;
    int lane = (int)(t & 31);
    if (e >= E) return;

    int s = (int)src[e];
    int d = (int)dst[e];
    float w = ew[e];

    const float4 xv = *(const float4*)(x + (size_t)s * 128 + lane * 4);
    float* ap = agg + (size_t)d * 128 + lane * 4;
    atomicAdd(ap + 0, xv.x * w);
    atomicAdd(ap + 1, xv.y * w);
    atomicAdd(ap + 2, xv.z * w);
    atomicAdd(ap + 3, xv.w * w);
    if (lane == 0) atomicAdd(deg + d, 1.0f);
}

// ---------------------------------------------------------------------------
// Kernel 2: fused dual GEMM + degree normalization using fp32 WMMA.
//   out[m, n] = (x[m,:] @ W0[:, n] + agg[m,:] @ W[:, n]) / max(deg[m], 1)
// One wave per 16x16 output tile. V_WMMA_F32_16X16X4_F32, K stepped by 4.
// Two independent accumulator chains (x*W0, agg*W) for WMMA pipelining.
//
// f32 16x4 A layout : lanes 0-15 -> M=lane, VGPR{0,1}=K{0,1};
//                     lanes 16-31 -> M=lane-16, VGPR{0,1}=K{2,3}
// f32 4x16 B layout : lanes 0-15 -> N=lane, VGPR{0,1}=K{0,1};
//                     lanes 16-31 -> N=lane-16, VGPR{0,1}=K{2,3}
// f32 16x16 C/D     : VGPR v -> M = v + 8*(lane>=16), N = lane&15
// ---------------------------------------------------------------------------
__global__ void __launch_bounds__(128)
gcn_gemm_kernel(const float* __restrict__ x,
                const float* __restrict__ agg,
                const float* __restrict__ deg,
                const float* __restrict__ W,
                const float* __restrict__ W0,
                float* __restrict__ out,
                int N) {
    const int wavesPerBlock = blockDim.x >> 5;
    const int wave = blockIdx.x * wavesPerBlock + ((int)threadIdx.x >> 5);
    const int lane = (int)threadIdx.x & 31;

    const int nTilesN = 128 / 16;                 // 8 column tiles
    const int mTiles  = (N + 15) / 16;
    const int mt = wave / nTilesN;
    const int nt = wave % nTilesN;
    if (mt >= mTiles) return;

    const int m0 = mt * 16;
    const int n0 = nt * 16;
    const int l    = lane & 15;                   // position within half
    const int half = lane >> 4;                   // 0: K={0,1}, 1: K={2,3}

    // Clamp row so all 32 lanes stay active (WMMA requires EXEC all-ones);
    // stores are predicated in the epilogue instead.
    int arow_m = m0 + l;
    if (arow_m >= N) arow_m = N - 1;

    const float* xrow = x   + (size_t)arow_m * 128 + 2 * half;   // A for chain 0
    const float* grow = agg + (size_t)arow_m * 128 + 2 * half;   // A for chain 1
    const float* w0c  = W0  + (size_t)(2 * half) * 128 + (n0 + l); // B chain 0
    const float* wc   = W   + (size_t)(2 * half) * 128 + (n0 + l); // B chain 1

    v8f c0 = {0.f, 0.f, 0.f, 0.f, 0.f, 0.f, 0.f, 0.f};
    v8f c1 = {0.f, 0.f, 0.f, 0.f, 0.f, 0.f, 0.f, 0.f};

#pragma unroll 8
    for (int k = 0; k < 128; k += 4) {
        // A fragments: two consecutive K elements per lane (8B aligned)
        v2f a0 = *(const v2f*)(xrow + k);
        v2f a1 = *(const v2f*)(grow + k);
        // B fragments: column-strided (coalesced across the 16 lanes)
        v2f b0; b0.x = w0c[(size_t)k * 128]; b0.y = w0c[(size_t)(k + 1) * 128];
        v2f b1; b1.x = wc [(size_t)k * 128]; b1.y = wc [(size_t)(k + 1) * 128];

        c0 = __builtin_amdgcn_wmma_f32_16x16x4_f32(
                 false, a0, false, b0, (short)0, c0, false, false);
        c1 = __builtin_amdgcn_wmma_f32_16x16x4_f32(
                 false, a1, false, b1, (short)0, c1, false, false);
    }

    // Epilogue: combine chains, divide by clamped in-degree, store.
#pragma unroll
    for (int v = 0; v < 8; ++v) {
        int row = m0 + v + 8 * half;
        if (row < N) {
            float dg  = deg[row];
            float inv = 1.0f / fmaxf(dg, 1.0f);
            out[(size_t)row * 128 + n0 + l] = (c0[v] + c1[v]) * inv;
        }
    }
}

// ---------------------------------------------------------------------------
// Launch: inputs are (x, edge_index[2,E] int64, num_nodes, edge_weight, W, W0)
// ---------------------------------------------------------------------------
extern "C" void kernel_launch(void* const* d_in, const int* in_sizes, int n_in,
                              void* d_out, int out_size, void* d_ws, size_t ws_size,
                              hipStream_t stream) {
    const float*     x    = (const float*)d_in[0];
    const long long* eidx = (const long long*)d_in[1];  // int64: [0..E)=src, [E..2E)=dst
    const float*     ew   = (const float*)d_in[3];
    const float*     W    = (const float*)d_in[4];
    const float*     W0   = (const float*)d_in[5];
    float*           out  = (float*)d_out;

    const int  N = in_sizes[0] / 128;
    const long long E = in_sizes[3];
    const long long* src = eidx;
    const long long* dst = eidx + E;

    float* agg = (float*)d_ws;                 // N*128 floats
    float* deg = agg + (size_t)N * 128;        // N floats (contiguous)

    // 1) zero agg + deg
    {
        long long n = (long long)N * 129;
        gcn_zero_kernel<<<4096, 256, 0, stream>>>(agg, n);
    }

    // 2) edge aggregation (one edge per 32 threads)
    {
        long long threads = E * 32;
        long long blocks = (threads + 255) / 256;
        gcn_aggregate_kernel<<<(unsigned)blocks, 256, 0, stream>>>(
            x, src, dst, ew, agg, deg, E);
    }

    // 3) fused dual GEMM + normalization (one wave per 16x16 tile)
    {
        int mTiles = (N + 15) / 16;
        long long tiles = (long long)mTiles * (128 / 16);
        int wavesPerBlock = 4;  // 128 threads
        long long blocks = (tiles + wavesPerBlock - 1) / wavesPerBlock;
        gcn_gemm_kernel<<<(unsigned)blocks, 32 * wavesPerBlock, 0, stream>>>(
            x, agg, deg, W, W0, out, N);
    }
}